// DCCFEncoder_46505905881231
// MI455X (gfx1250) — compile-verified
//
#include <hip/hip_runtime.h>
#include <math.h>

// Problem constants (match reference)
#define NU 100000
#define NI 50000
#define NN 150000       // NU + NI
#define DD 64
#define KK 128
#define LLAYERS 2
#define EE 1250000

typedef __attribute__((ext_vector_type(2))) float v2f;
typedef __attribute__((ext_vector_type(8))) float v8f;

// ---------------------------------------------------------------------------
// init: ek = concat(user_emb, item_emb); final = ek
// ---------------------------------------------------------------------------
__global__ __launch_bounds__(256)
void init_kernel(const float* __restrict__ ue, const float* __restrict__ ie,
                 float* __restrict__ ek, float* __restrict__ fin) {
  long long i = (long long)blockIdx.x * blockDim.x + threadIdx.x;
  const long long tot = (long long)NN * DD;
  if (i >= tot) return;
  const long long usz = (long long)NU * DD;
  float v = (i < usz) ? ue[i] : ie[i - usz];
  ek[i]  = v;
  fin[i] = v;
}

// ---------------------------------------------------------------------------
// SPMM: y[h[e]] += vals[e] * x[t[e]]   (one wave per edge, float2 per lane)
// ---------------------------------------------------------------------------
__global__ __launch_bounds__(256)
void spmm_kernel(const float* __restrict__ vals, const int* __restrict__ h,
                 const int* __restrict__ t, const float* __restrict__ x,
                 float* __restrict__ y) {
  int e = blockIdx.x * 8 + (threadIdx.x >> 5);
  if (e >= EE) return;
  int lane = threadIdx.x & 31;
  float v = vals[e];
  long long hh = h[e], tt = t[e];
  float2 d = ((const float2*)(x + tt * DD))[lane];
  float* dst = y + hh * DD + lane * 2;
  atomicAdd(dst + 0, v * d.x);
  atomicAdd(dst + 1, v * d.y);
}

// SPMM with adaptive values: v = alpha[e] * (rsum[h]>0 ? 1/rsum[h] : 0)
__global__ __launch_bounds__(256)
void spmm_scaled_kernel(const float* __restrict__ ealpha,
                        const float* __restrict__ rsum,
                        const int* __restrict__ h, const int* __restrict__ t,
                        const float* __restrict__ x, float* __restrict__ y) {
  int e = blockIdx.x * 8 + (threadIdx.x >> 5);
  if (e >= EE) return;
  int lane = threadIdx.x & 31;
  long long hh = h[e], tt = t[e];
  float rs = rsum[hh];
  float v = (rs > 0.0f) ? (ealpha[e] / rs) : 0.0f;
  float2 d = ((const float2*)(x + tt * DD))[lane];
  float* dst = y + hh * DD + lane * 2;
  atomicAdd(dst + 0, v * d.x);
  atomicAdd(dst + 1, v * d.y);
}

// ---------------------------------------------------------------------------
// per-node inverse L2 norm: invn[n] = 1 / max(||x[n]||, 1e-12)
// ---------------------------------------------------------------------------
__global__ __launch_bounds__(256)
void norm_kernel(const float* __restrict__ x, float* __restrict__ invn) {
  int n = blockIdx.x * 8 + (threadIdx.x >> 5);
  if (n >= NN) return;
  int lane = threadIdx.x & 31;
  float2 d = ((const float2*)(x + (long long)n * DD))[lane];
  float s = d.x * d.x + d.y * d.y;
  #pragma unroll
  for (int m = 16; m >= 1; m >>= 1) s += __shfl_xor(s, m, 32);
  if (lane == 0) invn[n] = 1.0f / fmaxf(sqrtf(s), 1e-12f);
}

// ---------------------------------------------------------------------------
// per-edge alpha = (cos(x[h],x[t]) + 1)/2; rsum[h] += alpha
// ---------------------------------------------------------------------------
__global__ __launch_bounds__(256)
void alpha_kernel(const float* __restrict__ x, const float* __restrict__ invn,
                  const int* __restrict__ h, const int* __restrict__ t,
                  float* __restrict__ ealpha, float* __restrict__ rsum) {
  int e = blockIdx.x * 8 + (threadIdx.x >> 5);
  if (e >= EE) return;
  int lane = threadIdx.x & 31;
  long long hh = h[e], tt = t[e];
  float2 a = ((const float2*)(x + hh * DD))[lane];
  float2 b = ((const float2*)(x + tt * DD))[lane];
  float s = a.x * b.x + a.y * b.y;
  #pragma unroll
  for (int m = 16; m >= 1; m >>= 1) s += __shfl_xor(s, m, 32);
  if (lane == 0) {
    float alpha = (s * invn[hh] * invn[tt] + 1.0f) * 0.5f;
    ealpha[e] = alpha;
    atomicAdd(&rsum[hh], alpha);
  }
}

// ---------------------------------------------------------------------------
// combine: ekn = gnn + intl + gaa + iaa + ek ; final += ekn
// ---------------------------------------------------------------------------
__global__ __launch_bounds__(256)
void combine_kernel(const float* __restrict__ g, const float* __restrict__ it,
                    const float* __restrict__ ga, const float* __restrict__ ia,
                    const float* __restrict__ ek, float* __restrict__ ekn,
                    float* __restrict__ fin) {
  long long i = (long long)blockIdx.x * blockDim.x + threadIdx.x;
  if (i >= (long long)NN * DD) return;
  float v = g[i] + it[i] + ga[i] + ia[i] + ek[i];
  ekn[i]  = v;
  fin[i] += v;
}

// ---------------------------------------------------------------------------
// Intent transform via fp32 WMMA:
//   out = softmax(E @ W, axis=1) @ W^T
//   E: rows x 64 (row-major), W: 64 x 128 (row-major), out: rows x 64
// One wave (32 threads) per 16-row tile. rows % 16 == 0.
//
// V_WMMA_F32_16X16X4_F32 operand layout (ISA 7.12.2, fp32 16x4 A):
//   lane<16:  row = lane,    a[v] = A[row][k0 + v]        (v = 0,1)
//   lane>=16: row = lane-16, a[v] = A[row][k0 + 2 + v]
//   B mirrored: col = lane&15, b[v] = B[k0 + half + v][col], half = lane<16?0:2
//   C/D: col = lane&15, c[r] = C[r + (lane<16?0:8)][col]
// ---------------------------------------------------------------------------
__global__ __launch_bounds__(32)
void intent_kernel(const float* __restrict__ emb, const float* __restrict__ W,
                   float* __restrict__ outp, int rows) {
  __shared__ float P[16 * KK];  // 8 KB: logits -> probs staging
  int row0 = blockIdx.x * 16;
  if (row0 >= rows) return;
  int lane = threadIdx.x;            // 0..31
  int m    = lane & 15;              // row (A) / col (B, C/D) index
  int half = (lane < 16) ? 0 : 2;    // K sub-offset per ISA layout
  int rb   = (lane < 16) ? 0 : 8;    // C/D row base

  // ---- GEMM1: logits[16x128] = E_tile[16x64] @ W[64x128] ----
  for (int nt = 0; nt < 8; ++nt) {
    int n0 = nt * 16;
    v8f c = {0.f, 0.f, 0.f, 0.f, 0.f, 0.f, 0.f, 0.f};
    #pragma unroll
    for (int k0 = 0; k0 < DD; k0 += 4) {
      v2f a, b;
      float2 av = *(const float2*)(emb + (long long)(row0 + m) * DD + k0 + half);
      a.x = av.x; a.y = av.y;
      b.x = W[(k0 + half + 0) * KK + n0 + m];
      b.y = W[(k0 + half + 1) * KK + n0 + m];
      c = __builtin_amdgcn_wmma_f32_16x16x4_f32(false, a, false, b,
                                                (short)0, c, false, false);
    }
    #pragma unroll
    for (int r = 0; r < 8; ++r) P[(rb + r) * KK + n0 + m] = c[r];
  }
  __syncthreads();

  // ---- row softmax over 128 cols (lanes 0..15, one row each) ----
  if (lane < 16) {
    float mx = -1e30f;
    for (int j = 0; j < KK; ++j) mx = fmaxf(mx, P[lane * KK + j]);
    float s = 0.0f;
    for (int j = 0; j < KK; ++j) {
      float e = expf(P[lane * KK + j] - mx);
      P[lane * KK + j] = e;
      s += e;
    }
    float inv = 1.0f / s;
    for (int j = 0; j < KK; ++j) P[lane * KK + j] *= inv;
  }
  __syncthreads();

  // ---- GEMM2: out[16x64] = P[16x128] @ W^T   (B[j][d] = W[d][j]) ----
  for (int nt = 0; nt < 4; ++nt) {
    int n0 = nt * 16;
    v8f c = {0.f, 0.f, 0.f, 0.f, 0.f, 0.f, 0.f, 0.f};
    #pragma unroll
    for (int k0 = 0; k0 < KK; k0 += 4) {
      v2f a, b;
      float2 av = *(const float2*)(&P[m * KK + k0 + half]);
      a.x = av.x; a.y = av.y;
      float2 bv = *(const float2*)(W + (n0 + m) * KK + k0 + half);
      b.x = bv.x; b.y = bv.y;
      c = __builtin_amdgcn_wmma_f32_16x16x4_f32(false, a, false, b,
                                                (short)0, c, false, false);
    }
    #pragma unroll
    for (int r = 0; r < 8; ++r)
      outp[(long long)(row0 + rb + r) * DD + n0 + m] = c[r];
  }
}

// ---------------------------------------------------------------------------
// Host orchestration
// ---------------------------------------------------------------------------
extern "C" void kernel_launch(void* const* d_in, const int* in_sizes, int n_in,
                              void* d_out, int out_size, void* d_ws, size_t ws_size,
                              hipStream_t stream) {
  (void)in_sizes; (void)n_in; (void)out_size; (void)ws_size;
  const float* user_emb  = (const float*)d_in[0];
  const float* item_emb  = (const float*)d_in[1];
  const float* user_intw = (const float*)d_in[2];
  const float* item_intw = (const float*)d_in[3];
  const float* adj_vals  = (const float*)d_in[4];
  const int*   h_idx     = (const int*)d_in[5];
  const int*   t_idx     = (const int*)d_in[6];
  // d_in[7]/d_in[8] (users/items) unused by the reference body.

  const size_t nd = (size_t)NN * DD;
  float* out      = (float*)d_out;
  float* fin      = out;                       // [ua ‖ ia] == final accumulator
  float* gnn_base = out + nd;                  // L x N x D
  float* int_base = gnn_base + (size_t)LLAYERS * nd;
  float* gaa_base = int_base + (size_t)LLAYERS * nd;
  float* iaa_base = gaa_base + (size_t)LLAYERS * nd;

  float* ws    = (float*)d_ws;
  float* ek    = ws;                            // N x D
  float* ek2   = ek + nd;                       // N x D
  float* ealpha = ek2 + nd;                     // E
  float* rsum   = ealpha + (size_t)EE;          // N
  float* invn   = rsum + (size_t)NN;            // N

  const int EDGE_BLKS = (EE + 7) / 8;
  const int NODE_BLKS = (NN + 7) / 8;
  const int ELEM_BLKS = (int)((nd + 255) / 256);

  init_kernel<<<ELEM_BLKS, 256, 0, stream>>>(user_emb, item_emb, ek, fin);

  for (int l = 0; l < LLAYERS; ++l) {
    float* gnn  = gnn_base + (size_t)l * nd;
    float* intl = int_base + (size_t)l * nd;
    float* gaa  = gaa_base + (size_t)l * nd;
    float* iaa  = iaa_base + (size_t)l * nd;

    // gnn = spmm(adj_vals, ek)
    hipMemsetAsync(gnn, 0, nd * sizeof(float), stream);
    spmm_kernel<<<EDGE_BLKS, 256, 0, stream>>>(adj_vals, h_idx, t_idx, ek, gnn);

    // intl = concat(intent(users), intent(items))
    intent_kernel<<<NU / 16, 32, 0, stream>>>(ek, user_intw, intl, NU);
    intent_kernel<<<NI / 16, 32, 0, stream>>>(ek + (size_t)NU * DD, item_intw,
                                              intl + (size_t)NU * DD, NI);

    // gaa = spmm(adaptive_vals(gnn[h], gnn[t]), ek)
    norm_kernel<<<NODE_BLKS, 256, 0, stream>>>(gnn, invn);
    hipMemsetAsync(rsum, 0, (size_t)NN * sizeof(float), stream);
    alpha_kernel<<<EDGE_BLKS, 256, 0, stream>>>(gnn, invn, h_idx, t_idx, ealpha, rsum);
    hipMemsetAsync(gaa, 0, nd * sizeof(float), stream);
    spmm_scaled_kernel<<<EDGE_BLKS, 256, 0, stream>>>(ealpha, rsum, h_idx, t_idx, ek, gaa);

    // iaa = spmm(adaptive_vals(intl[h], intl[t]), ek)
    norm_kernel<<<NODE_BLKS, 256, 0, stream>>>(intl, invn);
    hipMemsetAsync(rsum, 0, (size_t)NN * sizeof(float), stream);
    alpha_kernel<<<EDGE_BLKS, 256, 0, stream>>>(intl, invn, h_idx, t_idx, ealpha, rsum);
    hipMemsetAsync(iaa, 0, nd * sizeof(float), stream);
    spmm_scaled_kernel<<<EDGE_BLKS, 256, 0, stream>>>(ealpha, rsum, h_idx, t_idx, ek, iaa);

    // ek2 = gnn + intl + gaa + iaa + ek ; fin += ek2
    combine_kernel<<<ELEM_BLKS, 256, 0, stream>>>(gnn, intl, gaa, iaa, ek, ek2, fin);

    float* tmp = ek; ek = ek2; ek2 = tmp;  // ping-pong (deterministic per call)
  }
}